// Model_67551245631852
// MI455X (gfx1250) — compile-verified
//
#include <hip/hip_runtime.h>

// ---- problem constants (fixed by the reference harness) ----
#define B_   32
#define C_   256
#define H_   32
#define W_   128
#define A_   64
#define NPOS (A_ * A_)        // 4096 spatial positions
#define IMG  (H_ * W_)        // 4096 floats per (b,c) image
#define CCH  4                // channels staged per block -> 64 KB LDS
#define THREADS 256
#define POS_PER_THREAD (NPOS / THREADS)   // 16
#define SLAB_ELEMS (CCH * IMG)            // 16384 floats = 64 KB

#define PI_F 3.14159265358979323846f

// ---- CDNA5 staging-path feature detection ----
#if defined(__has_builtin)
#if __has_builtin(__builtin_amdgcn_tensor_load_to_lds)
#define HAVE_TDM 1
#endif
#if __has_builtin(__builtin_amdgcn_global_load_async_to_lds_b128)
#define HAVE_ASYNC_LDS 1
#endif
#endif
#ifndef HAVE_TDM
#define HAVE_TDM 0
#endif
#ifndef HAVE_ASYNC_LDS
#define HAVE_ASYNC_LDS 0
#endif

typedef int          vint4  __attribute__((ext_vector_type(4)));
typedef int          vint8  __attribute__((ext_vector_type(8)));
typedef unsigned int vuint4 __attribute__((ext_vector_type(4)));
typedef __attribute__((address_space(1))) vint4 gvint4;
typedef __attribute__((address_space(3))) vint4 lvint4;
typedef __attribute__((address_space(3))) float lfloat;

struct Bil {
    int   a00, a01, a10, a11;
    float w00, w01, w10, w11;
    float valid;
};

// Bilinear sampling parameters for homogeneous point (u1, v1, d1).
// Replicates the reference: uv = uv1[:2] / max(uv1[2], 1e-6), zero padding,
// valid mask from the unclipped coords, indices clipped (in float, to avoid
// int overflow for the huge UVs produced by the depth clamp).
__device__ __forceinline__ Bil make_bil(float u1, float v1, float d1) {
    float d  = fmaxf(d1, 1e-6f);
    float rd = __frcp_rn(d);
    float ix = u1 * rd;
    float iy = v1 * rd;
    float x0f = floorf(ix), y0f = floorf(iy);
    float fx = ix - x0f,    fy = iy - y0f;
    Bil r;
    r.w00 = (1.f - fx) * (1.f - fy);
    r.w01 = fx * (1.f - fy);
    r.w10 = (1.f - fx) * fy;
    r.w11 = fx * fy;
    r.valid = (ix >= 0.f && ix <= (float)(W_ - 1) &&
               iy >= 0.f && iy <= (float)(H_ - 1)) ? 1.f : 0.f;
    int x0 = (int)fminf(fmaxf(x0f,       0.f), (float)(W_ - 1));
    int x1 = (int)fminf(fmaxf(x0f + 1.f, 0.f), (float)(W_ - 1));
    int y0 = (int)fminf(fmaxf(y0f,       0.f), (float)(H_ - 1));
    int y1 = (int)fminf(fmaxf(y0f + 1.f, 0.f), (float)(H_ - 1));
    r.a00 = y0 * W_ + x0;
    r.a01 = y0 * W_ + x1;
    r.a10 = y1 * W_ + x0;
    r.a11 = y1 * W_ + x1;
    return r;
}

__global__ __launch_bounds__(THREADS)
void proj_blend_kernel(const float* __restrict__ grd,   // [B,C,H,W]
                       const float* __restrict__ phgt,  // [B,1,A,A]
                       const float* __restrict__ shu,   // [B,1]
                       const float* __restrict__ shv,   // [B,1]
                       const float* __restrict__ head,  // [B,1]
                       const float* __restrict__ cam,   // [B,3,3]
                       const float* __restrict__ sat,   // [1,C,A,A]
                       float* __restrict__ out)         // [B,C,A,A]
{
    __shared__ float smem[SLAB_ELEMS];   // 64 KB: CCH channels of grd[b]

    const int b   = blockIdx.y;
    const int c0  = blockIdx.x * CCH;
    const int tid = threadIdx.x;

    // ---- stage grd[b, c0:c0+CCH, :, :] (64 KB contiguous) into LDS ----
    const float* gsrc = grd + ((size_t)(b * C_ + c0)) * IMG;
#if HAVE_TDM
    if (tid < 32) {   // wave 0 issues one TDM DMA for the whole slab
        unsigned long long ga = (unsigned long long)(size_t)gsrc;
        unsigned int lds_off  = (unsigned int)(size_t)(lfloat*)smem;
        // D# group 0: count=1 | lds_addr | global_addr[56:0] | type=2
        vuint4 g0;
        g0.x = 1u;                                    // count=1, no gather
        g0.y = lds_off;                               // lds_addr [63:32]
        g0.z = (unsigned int)(ga & 0xFFFFFFFFull);    // global_addr[31:0]
        g0.w = (unsigned int)((ga >> 32) & 0x01FFFFFFull) | (2u << 30); // [56:32]|type=2
        // D# group 1: data_size=4B, tensor/tile dim0=16384, dim1=1, stride=16384
        vint8 g1;
        g1[0] = 0x00020000;                  // workgroup_mask=0, data_size=2 (4B)
        g1[1] = (int)(SLAB_ELEMS << 16);     // tensor_dim0[15:0] @ bits 63:48
        g1[2] = (int)(1 << 16);              // tensor_dim0[31:16]=0, tensor_dim1=1
        g1[3] = (int)(SLAB_ELEMS << 16);     // tensor_dim1[31:16]=0, tile_dim0
        g1[4] = 1;                           // tile_dim1=1, tile_dim2=0
        g1[5] = SLAB_ELEMS;                  // tensor_dim0_stride[31:0]
        g1[6] = 0;                           // stride[47:32]=0, dim1_stride lo=0
        g1[7] = 0;
        vint4 gz = {0, 0, 0, 0};             // groups 2/3 unused (2D tensor)
#if __clang_major__ >= 23
        vint8 gz8 = {0, 0, 0, 0, 0, 0, 0, 0};
        __builtin_amdgcn_tensor_load_to_lds(g0, g1, gz, gz, gz8, 0);
#else
        __builtin_amdgcn_tensor_load_to_lds(g0, g1, gz, gz, 0);
#endif
#if defined(__has_builtin) && __has_builtin(__builtin_amdgcn_s_wait_tensorcnt)
        __builtin_amdgcn_s_wait_tensorcnt(0);
#else
        asm volatile("s_wait_tensorcnt 0x0" ::: "memory");
#endif
    }
#elif HAVE_ASYNC_LDS
    {
        const int nvec = SLAB_ELEMS / 4;   // float4 chunks
        for (int i = tid; i < nvec; i += THREADS) {
            __builtin_amdgcn_global_load_async_to_lds_b128(
                (gvint4*)(gsrc + i * 4),
                (lvint4*)(smem + i * 4),
                0, 0);
        }
#if defined(__has_builtin) && __has_builtin(__builtin_amdgcn_s_wait_asynccnt)
        __builtin_amdgcn_s_wait_asynccnt(0);
#else
        asm volatile("s_wait_asynccnt 0" ::: "memory");
#endif
    }
#else
    {
        const int nvec = SLAB_ELEMS / 4;
        const float4* g4 = (const float4*)gsrc;
        float4*       s4 = (float4*)smem;
        for (int i = tid; i < nvec; i += THREADS) s4[i] = g4[i];
    }
#endif

    // ---- per-batch projection matrix P = K_scaled @ [R | T] (uniform) ----
    // (waves 1-7 compute this while the DMA is in flight; wave 0 waits first)
    float ang = -head[b] * (10.0f / 180.0f) * PI_F;
    float cs = cosf(ang), sn = sinf(ang);
    float su = 20.0f * shu[b];
    float sv = 20.0f * shv[b];
    float Tx = sv, Ty = 1.65f, Tz = -su;
    const float sx = (float)W_ / 1024.0f;   // W / ORI_W
    const float sy = (float)H_ / 256.0f;    // H / ORI_H
    const float* Kb = cam + b * 9;
    float K00 = Kb[0] * sx, K01 = Kb[1] * sx, K02 = Kb[2] * sx;
    float K10 = Kb[3] * sy, K11 = Kb[4] * sy, K12 = Kb[5] * sy;
    float K20 = Kb[6],      K21 = Kb[7],      K22 = Kb[8];
    // R columns: (cs,0,sn), (0,1,0), (-sn,0,cs)
    float P00 = K00 * cs + K02 * sn, P01 = K01, P02 = -K00 * sn + K02 * cs;
    float P03 = K00 * Tx + K01 * Ty + K02 * Tz;
    float P10 = K10 * cs + K12 * sn, P11 = K11, P12 = -K10 * sn + K12 * cs;
    float P13 = K10 * Tx + K11 * Ty + K12 * Tz;
    float P20 = K20 * cs + K22 * sn, P21 = K21, P22 = -K20 * sn + K22 * cs;
    float P23 = K20 * Tx + K21 * Ty + K22 * Tz;

    __syncthreads();   // slab visible to all waves

    const float mpp = 0.2f * 512.0f / (float)A_;   // 1.6
    const float ctr = (float)(A_ / 2);

    for (int k = 0; k < POS_PER_THREAD; ++k) {
        int p  = tid + k * THREADS;            // consecutive lanes -> consecutive w
        int hh = p >> 6;                       // A_ == 64
        int ww = p & (A_ - 1);
        float X = mpp * ((float)hh - ctr);
        float Z = mpp * ((float)ww - ctr);

        // flat-ground projection (Y = 0)
        float u1 = P00 * X + P02 * Z + P03;
        float v1 = P10 * X + P12 * Z + P13;
        float d1 = P20 * X + P22 * Z + P23;
        // height-aware projection: Y = -height  ->  uv1_h = uv1 - P[:,1]*h
        float hg  = phgt[b * NPOS + p];
        float u1h = u1 - P01 * hg;
        float v1h = v1 - P11 * hg;
        float d1h = d1 - P21 * hg;

        Bil b0 = make_bil(u1,  v1,  d1);
        Bil bh = make_bil(u1h, v1h, d1h);

#pragma unroll
        for (int c = 0; c < CCH; ++c) {
            const float* s = smem + c * IMG;
            float g0v = b0.w00 * s[b0.a00] + b0.w01 * s[b0.a01] +
                        b0.w10 * s[b0.a10] + b0.w11 * s[b0.a11];
            g0v *= b0.valid;
            float ghv = bh.w00 * s[bh.a00] + bh.w01 * s[bh.a01] +
                        bh.w10 * s[bh.a10] + bh.w11 * s[bh.a11];
            ghv *= bh.valid;
            float gate = sat[(c0 + c) * NPOS + p];          // coalesced, L2-hot
            float rv   = ghv * gate + g0v * (1.0f - gate);
            // streamed output, no reuse -> non-temporal store
            __builtin_nontemporal_store(
                rv, out + ((size_t)(b * C_ + c0 + c)) * NPOS + p);
        }
    }
}

extern "C" void kernel_launch(void* const* d_in, const int* in_sizes, int n_in,
                              void* d_out, int out_size, void* d_ws, size_t ws_size,
                              hipStream_t stream) {
    const float* grd  = (const float*)d_in[0];  // grd_f          [B,C,H,W]
    const float* phg  = (const float*)d_in[1];  // predict_height [B,1,A,A]
    const float* shu  = (const float*)d_in[2];  // shift_u        [B,1]
    const float* shv  = (const float*)d_in[3];  // shift_v        [B,1]
    const float* hea  = (const float*)d_in[4];  // heading        [B,1]
    const float* cam  = (const float*)d_in[5];  // camera_k       [B,3,3]
    const float* sat  = (const float*)d_in[6];  // sat_embedding  [1,C,A,A]
    (void)in_sizes; (void)n_in; (void)out_size; (void)d_ws; (void)ws_size;

    dim3 grid(C_ / CCH, B_);   // (64, 32)
    proj_blend_kernel<<<grid, THREADS, 0, stream>>>(
        grd, phg, shu, shv, hea, cam, sat, (float*)d_out);
}